// Tsallis15Loss_12421045420952
// MI455X (gfx1250) — compile-verified
//
#include <hip/hip_runtime.h>
#include <stdint.h>

// Tsallis-1.5 (entmax-1.5) loss for [2048, 32000] fp32 logits on MI455X.
//
// Sort-free: tau* is the root of f(tau) = sum relu((x-max)/2 - tau)^2 - 1
// on [-1, 0]. One 1024-thread workgroup per row:
//   1) async-copy the 128 KB row global->LDS (GLOBAL_LOAD_ASYNC_TO_LDS_B128)
//   2) 32 VGPRs/thread; 6 coarse bisection iterations (bracket -> 1/64)
//   3) elements with s <= lo can never re-enter the support -> deterministic
//      block-scan compaction of actives into LDS (typically << 512 of 32000)
//   4) fine phase: 24 bisection iters + reference's closed-form refinement
//      + final sums, done by a single wave in registers (no barriers) when
//      cnt <= 512, else by the whole block over the compacted LDS array.
// Deterministic final reduction in a second single-block kernel (no atomics).

#define ROW_D      32000
#define THREADS    1024
#define EPT        32
#define SENT       (-1.0e30f)
#define FULL_ITERS 6
#define FINE_ITERS 24
#define SMALL_CAP  512          // 16 regs/lane * 32 lanes

// 16-byte vector type matching the async-LDS builtin's expected pointee.
typedef int v4i __attribute__((vector_size(16)));
typedef __attribute__((address_space(1))) v4i* gptr_v4i;   // global
typedef __attribute__((address_space(3))) v4i* lptr_v4i;   // LDS

__device__ __forceinline__ float wsum(float v) {
#pragma unroll
  for (int o = 16; o; o >>= 1) v += __shfl_xor(v, o, 32);
  return v;
}

__device__ __forceinline__ float wmaxr(float v) {
#pragma unroll
  for (int o = 16; o; o >>= 1) v = fmaxf(v, __shfl_xor(v, o, 32));
  return v;
}

// blockDim.x must be 1024 (exactly 32 waves). fred needs 33 floats.
__device__ __forceinline__ float block_sum(float a, float* fred) {
  const int lane = threadIdx.x & 31;
  const int wave = threadIdx.x >> 5;
  a = wsum(a);
  if (lane == 0) fred[wave] = a;
  __syncthreads();
  if (wave == 0) {
    float t = fred[lane];
    t = wsum(t);
    if (lane == 0) fred[32] = t;
  }
  __syncthreads();
  float r = fred[32];
  __syncthreads();
  return r;
}

__device__ __forceinline__ float block_max(float a, float* fred) {
  const int lane = threadIdx.x & 31;
  const int wave = threadIdx.x >> 5;
  a = wmaxr(a);
  if (lane == 0) fred[wave] = a;
  __syncthreads();
  if (wave == 0) {
    float t = fred[lane];
    t = wmaxr(t);
    if (lane == 0) fred[32] = t;
  }
  __syncthreads();
  float r = fred[32];
  __syncthreads();
  return r;
}

__global__ __launch_bounds__(THREADS)
void tsallis15_row_kernel(const float* __restrict__ x,
                          const long long* __restrict__ tgt,
                          float* __restrict__ row_loss)
{
  extern __shared__ char smem_raw[];
  float* row  = (float*)smem_raw;                            // ROW_D floats (later: compacted actives)
  float* fred = (float*)(smem_raw + ROW_D * sizeof(float));  // 34 floats
  int*   ired = (int*)(fred + 34);                           // 66 ints

  const int tid  = threadIdx.x;
  const int lane = tid & 31;
  const int wave = tid >> 5;
  const int r    = blockIdx.x;
  const float* src = x + (size_t)r * ROW_D;

  // ---- 1) stage row into LDS via gfx1250 async global->LDS copies ----
#if __has_builtin(__builtin_amdgcn_global_load_async_to_lds_b128)
  for (int i = tid; i < ROW_D / 4; i += THREADS) {
    __builtin_amdgcn_global_load_async_to_lds_b128(
        (gptr_v4i)(void*)(src + 4 * i),
        (lptr_v4i)(row + 4 * i),
        0, 0);
  }
#if __has_builtin(__builtin_amdgcn_s_wait_asynccnt)
  __builtin_amdgcn_s_wait_asynccnt(0);
#else
  asm volatile("s_wait_asynccnt 0" ::: "memory");
#endif
#else
  for (int i = tid; i < ROW_D / 4; i += THREADS) {
    ((float4*)row)[i] = ((const float4*)src)[i];
  }
#endif
  __syncthreads();

  // ---- 2) LDS -> registers (strided: conflict-free, coalesced per slot) ----
  float v[EPT];
#pragma unroll
  for (int k = 0; k < EPT; ++k) {
    int j = tid + k * THREADS;
    v[k] = (j < ROW_D) ? row[j] : SENT;
  }

  // target logit for this row (read before row[] is reused for compaction;
  // barriers inside the coarse bisection order this write-after-read safely)
  float xt = 0.0f;
  if (tid == 0) {
    long long t = tgt[r];
    if (t >= 0 && t < ROW_D) xt = row[(int)t];
  }

  // ---- 3) row max, then s = (x - m) * 0.5 in-place ----
  float m = SENT;
#pragma unroll
  for (int k = 0; k < EPT; ++k) m = fmaxf(m, v[k]);
  m = block_max(m, fred);

#pragma unroll
  for (int k = 0; k < EPT; ++k) v[k] = (v[k] - m) * 0.5f;
  // padded slots ~ -5e29: relu(s - tau) == 0 for tau in [-1,0], harmless.

  // ---- 4) coarse bisection: f(tau) = sum relu(s - tau)^2 over [-1, 0] ----
  float lo = -1.0f, hi = 0.0f;
  for (int it = 0; it < FULL_ITERS; ++it) {
    float tau = 0.5f * (lo + hi);
    float acc = 0.0f;
#pragma unroll
    for (int k = 0; k < EPT; ++k) {
      float y = fmaxf(v[k] - tau, 0.0f);
      acc = fmaf(y, y, acc);
    }
    acc = block_sum(acc, fred);          // uniform across block
    if (acc >= 1.0f) lo = tau; else hi = tau;
  }
  // Every element with s <= lo contributes 0 for all remaining tau >= lo.

  // ---- 5) deterministic compaction of actives {s > lo} into row[] ----
  int c = 0;
#pragma unroll
  for (int k = 0; k < EPT; ++k) c += (v[k] > lo) ? 1 : 0;

  int incl = c;                           // wave-level inclusive scan
#pragma unroll
  for (int o = 1; o < 32; o <<= 1) {
    int n = __shfl_up(incl, o, 32);
    if (lane >= o) incl += n;
  }
  if (lane == 31) ired[wave] = incl;      // wave totals
  __syncthreads();
  if (wave == 0) {
    int t0 = ired[lane];
    int ti = t0;
#pragma unroll
    for (int o = 1; o < 32; o <<= 1) {
      int n = __shfl_up(ti, o, 32);
      if (lane >= o) ti += n;
    }
    ired[32 + lane] = ti - t0;            // exclusive base per wave
    if (lane == 31) ired[64] = ti;        // total active count
  }
  __syncthreads();
  int base = ired[32 + wave] + (incl - c);
  const int cnt = ired[64];               // >= 1 (s_max = 0 > lo)

  int idx = base;
#pragma unroll
  for (int k = 0; k < EPT; ++k) {
    if (v[k] > lo) row[idx++] = v[k];
  }
  __syncthreads();

  // ---- 6) fine phase on the compacted set ----
  if (cnt <= SMALL_CAP) {
    // single wave, registers only, zero barriers
    if (wave == 0) {
      float w[16];
      const float pad = lo - 1.0f;        // < lo <= any tau: contributes 0
#pragma unroll
      for (int q = 0; q < 16; ++q) {
        int j = lane + 32 * q;
        w[q] = (j < cnt) ? row[j] : pad;
      }
      float l = lo, h = hi;
      for (int it = 0; it < FINE_ITERS; ++it) {
        float t = 0.5f * (l + h);
        float acc = 0.0f;
#pragma unroll
        for (int q = 0; q < 16; ++q) {
          float y = fmaxf(w[q] - t, 0.0f);
          acc = fmaf(y, y, acc);
        }
        acc = wsum(acc);
        if (acc >= 1.0f) l = t; else h = t;
      }
      float t0 = 0.5f * (l + h);
      float cf = 0.0f, s1 = 0.0f, s2 = 0.0f;
#pragma unroll
      for (int q = 0; q < 16; ++q) {
        if (w[q] > t0) { cf += 1.0f; s1 += w[q]; s2 = fmaf(w[q], w[q], s2); }
      }
      cf = wsum(cf); s1 = wsum(s1); s2 = wsum(s2);
      float mv  = s1 / cf;
      float tau = mv - sqrtf(fmaxf((1.0f - (s2 - s1 * mv)) / cf, 0.0f));

      float sy3 = 0.0f, sy2s = 0.0f, sy2 = 0.0f;
#pragma unroll
      for (int q = 0; q < 16; ++q) {
        float y  = fmaxf(w[q] - tau, 0.0f);
        float y2 = y * y;
        sy3  = fmaf(y2, y, sy3);
        sy2s = fmaf(y2, w[q], sy2s);
        sy2 += y2;
      }
      sy3 = wsum(sy3); sy2s = wsum(sy2s); sy2 = wsum(sy2);
      if (lane == 0) {
        float dot = 2.0f * sy2s + m * sy2;     // <p, x_original>
        row_loss[r] = (1.0f - sy3) / 0.75f + dot - xt;
      }
    }
    return;                                    // uniform exit
  }

  // fallback: whole block over the compacted LDS array
  {
    float l = lo, h = hi;
    for (int it = 0; it < FINE_ITERS; ++it) {
      float t = 0.5f * (l + h);
      float acc = 0.0f;
      for (int i = tid; i < cnt; i += THREADS) {
        float y = fmaxf(row[i] - t, 0.0f);
        acc = fmaf(y, y, acc);
      }
      acc = block_sum(acc, fred);
      if (acc >= 1.0f) l = t; else h = t;
    }
    float t0 = 0.5f * (l + h);
    float cf = 0.0f, s1 = 0.0f, s2 = 0.0f;
    for (int i = tid; i < cnt; i += THREADS) {
      float w = row[i];
      if (w > t0) { cf += 1.0f; s1 += w; s2 = fmaf(w, w, s2); }
    }
    cf = block_sum(cf, fred); s1 = block_sum(s1, fred); s2 = block_sum(s2, fred);
    float mv  = s1 / cf;
    float tau = mv - sqrtf(fmaxf((1.0f - (s2 - s1 * mv)) / cf, 0.0f));

    float sy3 = 0.0f, sy2s = 0.0f, sy2 = 0.0f;
    for (int i = tid; i < cnt; i += THREADS) {
      float w  = row[i];
      float y  = fmaxf(w - tau, 0.0f);
      float y2 = y * y;
      sy3  = fmaf(y2, y, sy3);
      sy2s = fmaf(y2, w, sy2s);
      sy2 += y2;
    }
    sy3  = block_sum(sy3, fred);
    sy2s = block_sum(sy2s, fred);
    sy2  = block_sum(sy2, fred);
    if (tid == 0) {
      float dot = 2.0f * sy2s + m * sy2;
      row_loss[r] = (1.0f - sy3) / 0.75f + dot - xt;
    }
  }
}

__global__ __launch_bounds__(THREADS)
void tsallis15_reduce_kernel(const float* __restrict__ row_loss,
                             float* __restrict__ out, int n)
{
  __shared__ float fred[34];
  const int tid = threadIdx.x;
  float a = 0.0f;
  for (int i = tid; i < n; i += THREADS) a += row_loss[i];
  a = block_sum(a, fred);
  if (tid == 0) out[0] = a / (float)n;
}

extern "C" void kernel_launch(void* const* d_in, const int* in_sizes, int n_in,
                              void* d_out, int out_size, void* d_ws, size_t ws_size,
                              hipStream_t stream) {
  const float*     inp = (const float*)d_in[0];      // [2048, 32000] fp32
  const long long* tgt = (const long long*)d_in[1];  // [2048] int64
  float* out = (float*)d_out;                        // scalar fp32
  float* row_loss = (float*)d_ws;                    // 2048 floats scratch

  const int n_rows = in_sizes[1];                    // 2048
  const size_t lds_bytes = (size_t)ROW_D * sizeof(float)
                         + 34 * sizeof(float) + 66 * sizeof(int);

  // Row buffer exceeds the 64 KB legacy default; raise the dynamic-LDS cap.
  (void)hipFuncSetAttribute((const void*)tsallis15_row_kernel,
                            hipFuncAttributeMaxDynamicSharedMemorySize,
                            (int)lds_bytes);

  tsallis15_row_kernel<<<n_rows, THREADS, lds_bytes, stream>>>(inp, tgt, row_loss);
  tsallis15_reduce_kernel<<<1, THREADS, 0, stream>>>(row_loss, out, n_rows);
}